// Simplified_MultiHeadAttention_78752520339640
// MI455X (gfx1250) — compile-verified
//
#include <hip/hip_runtime.h>
#include <stdint.h>

typedef float v4f __attribute__((ext_vector_type(4)));

namespace {
constexpr int Bn = 8;
constexpr int Tn = 4096;
constexpr int Dn = 512;
constexpr int TM = 16;                           // output rows per block
constexpr int HALO = 5;                          // deepest tap: X[m-5]
constexpr int LROWS = TM + HALO;                 // 21 staged rows
constexpr int C4 = Dn / 4;                       // 128 16B chunks per row
constexpr int NTILES = (Tn + 1 + TM - 1) / TM;   // 257 tiles over T+1 rows
constexpr int THREADS = 256;
constexpr int DOUT = 4 * Dn;                     // 2048 output floats per row
}

static __device__ __forceinline__ v4f vzero() {
  v4f z; z.x = 0.f; z.y = 0.f; z.z = 0.f; z.w = 0.f; return z;
}

__global__ __launch_bounds__(THREADS)
void band_attn_kernel(const float* __restrict__ X,
                      const float* __restrict__ W,
                      float* __restrict__ out) {
  __shared__ v4f tile[LROWS * C4];               // 21 * 512 floats = 42 KB

  const int blk = blockIdx.x;
  const int b   = blk / NTILES;
  const int m0  = (blk % NTILES) * TM;
  const int tid = threadIdx.x;
  const int c0  = tid & (C4 - 1);                // loop-invariant column chunk
  const int r0  = tid >> 7;                      // 0 or 1; rows step by 2

  // ---- Stage X rows [m0-5, m0+TM-1] (clamped to [0,Tn)) into LDS via
  // ---- CDNA5 async DMA (ASYNCcnt-tracked, no VGPR staging).
  {
    const unsigned ldsBase = (unsigned)(uintptr_t)&tile[0]; // low 32 bits = LDS offset
    unsigned la = ldsBase + (unsigned)((r0 * C4 + c0) * sizeof(v4f));
    const float* gp = X + (((size_t)b * Tn + (m0 - HALO + r0)) * Dn + (size_t)c0 * 4);
#pragma unroll
    for (int j = 0; j < (LROWS + 1) / 2; ++j) {  // 11 strided issues, stride 2 rows
      const int lr = r0 + 2 * j;
      const int g  = m0 - HALO + lr;
      if (lr < LROWS && g >= 0 && g < Tn) {
        asm volatile("global_load_async_to_lds_b128 %0, %1, off"
                     :: "v"(la), "v"(gp) : "memory");
      }
      la += 2u * C4 * (unsigned)sizeof(v4f);     // +2 rows in LDS
      gp += 2 * Dn;                               // +2 rows in global
    }
  }
  asm volatile("s_wait_asynccnt 0" ::: "memory");
  __syncthreads();

  // ---- Band-softmax weights from the real W input (uniform over interior rows).
  const float w00 = W[0 * 2 + 0], w10 = W[1 * 2 + 0];   // head 0 band values
  const float w11 = W[1 * 2 + 1], w21 = W[2 * 2 + 1];   // head 1 band values
  float mx = fmaxf(w00, w10);
  float e0 = expf(w00 - mx), e1 = expf(w10 - mx);
  const float a0 = e0 / (e0 + e1), a1 = e1 / (e0 + e1);
  mx = fmaxf(w11, w21);
  e0 = expf(w11 - mx); e1 = expf(w21 - mx);
  const float b0 = e0 / (e0 + e1), b1 = e1 / (e0 + e1);
  const float c3 = a0 * b0, c4 = a1 * b0 + a0 * b1, c5 = a1 * b1;  // seg3 taps

  const bool interior = (m0 >= HALO) && (m0 + TM <= Tn);  // 255 of 257 tiles

  if (interior) {
    // Branch-free fast path: all taps valid, all m < Tn.
    const v4f* lp = &tile[(r0 + HALO) * C4 + c0];          // tap-0 row
    float* op = out + (((size_t)b * (Tn + 1) + (m0 + r0)) * (size_t)DOUT
                       + (size_t)c0 * 4);
#pragma unroll
    for (int i = 0; i < TM * C4 / THREADS; ++i) {          // 8 iterations
      const v4f x0 = lp[0];
      const v4f x1 = lp[-1 * C4];
      const v4f x2 = lp[-2 * C4];
      const v4f x3 = lp[-3 * C4];
      const v4f x4 = lp[-4 * C4];
      const v4f x5 = lp[-5 * C4];
      const v4f s1 = a0 * x1 + a1 * x2;
      const v4f s2 = b0 * x2 + b1 * x3;
      const v4f s3 = c3 * x3 + c4 * x4 + c5 * x5;
      // 268 MB write stream, never re-read: non-temporal keeps X hot in L2.
      __builtin_nontemporal_store(x0, (v4f*)(op));
      __builtin_nontemporal_store(s1, (v4f*)(op + Dn));
      __builtin_nontemporal_store(s2, (v4f*)(op + 2 * Dn));
      __builtin_nontemporal_store(s3, (v4f*)(op + 3 * Dn));
      lp += 2 * C4;                 // +2 rows
      op += 2 * (size_t)DOUT;
    }
  } else {
    // Edge tiles: t==0 (softmax edge rows m<5) and t==256 (single row m==Tn).
    const int rowsHere = min(TM, Tn + 1 - m0);
    for (int u = tid; u < rowsHere * C4; u += THREADS) {
      const int r = u >> 7;
      const int c = u & (C4 - 1);
      const int m = m0 + r;

      auto tap = [&](int k) -> v4f {
        const int g = m - k;
        if (g < 0 || g >= Tn) return vzero();   // zero-pad: never read garbage LDS
        return tile[(r - k + HALO) * C4 + c];
      };

      const v4f x0 = tap(0), x1 = tap(1), x2 = tap(2);
      const v4f x3 = tap(3), x4 = tap(4), x5 = tap(5);

      v4f s1;                                   // seg1 = v0[m]
      if (m >= 2)      s1 = a0 * x1 + a1 * x2;
      else if (m == 1) s1 = x1;                 // single-entry row -> softmax == 1
      else             s1 = vzero();

      v4f s2;                                   // seg2 = v1[m][0:D]
      if (m >= 3)      s2 = b0 * x2 + b1 * x3;
      else if (m == 2) s2 = x2;
      else             s2 = vzero();

      v4f s3;                                   // seg3 = b0*v0[m-2] + b1*v0[m-3]
      if (m >= 5)      s3 = c3 * x3 + c4 * x4 + c5 * x5;
      else if (m == 4) s3 = c3 * x3 + (a1 * b0 + b1) * x4;  // v0[1]=X[0]
      else if (m == 3) s3 = b0 * x3;                         // v0[0]=0
      else             s3 = vzero();

      float* op = out + (((size_t)b * (Tn + 1) + m) * (size_t)DOUT + (size_t)c * 4);
      __builtin_nontemporal_store(x0, (v4f*)(op));
      __builtin_nontemporal_store(s1, (v4f*)(op + Dn));
      __builtin_nontemporal_store(s2, (v4f*)(op + 2 * Dn));
      __builtin_nontemporal_store(s3, (v4f*)(op + 3 * Dn));
    }
  }
}

extern "C" void kernel_launch(void* const* d_in, const int* in_sizes, int n_in,
                              void* d_out, int out_size, void* d_ws, size_t ws_size,
                              hipStream_t stream) {
  (void)in_sizes; (void)n_in; (void)out_size; (void)d_ws; (void)ws_size;
  const float* X = (const float*)d_in[0];
  const float* W = (const float*)d_in[1];
  float* out = (float*)d_out;
  band_attn_kernel<<<dim3(Bn * NTILES), dim3(THREADS), 0, stream>>>(X, W, out);
}